// RoutingLayer_20332375180095
// MI455X (gfx1250) — compile-verified
//
#include <hip/hip_runtime.h>
#include <stdint.h>

typedef __attribute__((ext_vector_type(2))) float v2f;
typedef __attribute__((ext_vector_type(8))) float v8f;

#define NROWS 100000
#define DIM   1024
#define ROUTE 64
#define ROWS_PER_BLOCK 128                                   // 8 waves * 16 rows
#define SCORE_BLOCKS ((NROWS + ROWS_PER_BLOCK - 1) / ROWS_PER_BLOCK)  // 782
#define CHUNKS 200
#define ROWS_PER_CHUNK (NROWS / CHUNKS)                      // 500

// ---- workspace layout (bytes); total ~1.22 MB ----
#define TAKEN_OFF   0u                         // 100000 f32
#define S_OFF       409600u                    // 1024 f32 col-sum
#define MEAN_OFF    (S_OFF + 4096u)            // 1024 f32
#define C_OFF       (MEAN_OFF + 4096u)         // 1 f32 (live count)
#define PART_OFF    (C_OFF + 256u)             // 200*1024 f32 partial colsums
#define BSCORE_OFF  (PART_OFF + CHUNKS*DIM*4u) // 782 f32
#define BIDX_OFF    (BSCORE_OFF + 3200u)       // 782 i32
#define IDXS_OFF    (BIDX_OFF + 3200u)         // 64 i32
#define SORT_OFF    (IDXS_OFF + 256u)          // 64 i32

__device__ __forceinline__ v8f wmma_f32_4(v2f a, v2f b, v8f c) {
  // V_WMMA_F32_16X16X4_F32 : D(16x16 f32) = A(16x4) * B(4x16) + C
  return __builtin_amdgcn_wmma_f32_16x16x4_f32(
      false, a, false, b, (short)0, c, false, false);
}

__global__ __launch_bounds__(256) void init_taken(float* __restrict__ taken) {
  int i = blockIdx.x * 256 + threadIdx.x;
  if (i < NROWS) taken[i] = 0.0f;
}

// Partial column-sums: block = (chunk, colblock). Coalesced column reads.
__global__ __launch_bounds__(256) void colsum_partial(const float* __restrict__ x,
                                                      float* __restrict__ part) {
  int cblk  = blockIdx.x & 3;
  int chunk = blockIdx.x >> 2;
  int col = cblk * 256 + threadIdx.x;
  const float* p = x + (size_t)chunk * ROWS_PER_CHUNK * DIM + col;
  float s = 0.0f;
  for (int r = 0; r < ROWS_PER_CHUNK; ++r) s += p[(size_t)r * DIM];
  part[(size_t)chunk * DIM + col] = s;
}

__global__ __launch_bounds__(256) void colsum_final(const float* __restrict__ part,
                                                    float* __restrict__ S,
                                                    float* __restrict__ mean,
                                                    float* __restrict__ c) {
  int col = blockIdx.x * 256 + threadIdx.x;
  float s = 0.0f;
  for (int ch = 0; ch < CHUNKS; ++ch) s += part[(size_t)ch * DIM + col];
  S[col] = s;
  mean[col] = s / (float)NROWS;
  if (col == 0) *c = (float)NROWS;
}

// Per-round scores via f32 WMMA: each wave computes 16 row-dots against mean.
// score_i = 1/(x_i . mean) + taken_i ; block-local argmax (first index on ties).
__global__ __launch_bounds__(256) void score_kernel(const float* __restrict__ x,
                                                    const float* __restrict__ mean,
                                                    const float* __restrict__ taken,
                                                    float* __restrict__ blkScore,
                                                    int*   __restrict__ blkSel) {
  __shared__ float smean[DIM];
  __shared__ float sSc[8];
  __shared__ int   sIx[8];
  for (int j = threadIdx.x; j < DIM; j += 256) smean[j] = mean[j];
  __syncthreads();

  const int wave = threadIdx.x >> 5;
  const int lane = threadIdx.x & 31;
  const int r0 = blockIdx.x * ROWS_PER_BLOCK + wave * 16;

  float bs = -__builtin_inff();
  int   bi = 0x7fffffff;

  if (r0 < NROWS) {                       // wave-uniform: EXEC all-1s for WMMA
    const int mrow  = lane & 15;          // A layout: M = lane%16
    const int khalf = (lane >> 4) << 1;   // lanes 16-31 hold K=2,3 slots
    const float* rowPtr = x + (size_t)(r0 + mrow) * DIM + khalf;
    const float* mPtr   = smean + khalf;  // B broadcast along N, same k slots

    v8f c0 = {}, c1 = {}, c2 = {}, c3 = {};
    for (int k = 0; k < DIM; k += 16) {   // 4 independent WMMA chains (ILP)
      v2f a0 = *(const v2f*)(rowPtr + k);      v2f b0 = *(const v2f*)(mPtr + k);
      v2f a1 = *(const v2f*)(rowPtr + k + 4);  v2f b1 = *(const v2f*)(mPtr + k + 4);
      v2f a2 = *(const v2f*)(rowPtr + k + 8);  v2f b2 = *(const v2f*)(mPtr + k + 8);
      v2f a3 = *(const v2f*)(rowPtr + k + 12); v2f b3 = *(const v2f*)(mPtr + k + 12);
      c0 = wmma_f32_4(a0, b0, c0);
      c1 = wmma_f32_4(a1, b1, c1);
      c2 = wmma_f32_4(a2, b2, c2);
      c3 = wmma_f32_4(a3, b3, c3);
    }
    v8f tot = (c0 + c1) + (c2 + c3);      // every D column holds the dots

    const int rbase = r0 + ((lane >> 4) << 3);  // C/D layout: lanes>=16 -> M=8..15
#pragma unroll
    for (int j = 0; j < 8; ++j) {
      int row = rbase + j;
      float sc = 1.0f / tot[j] + taken[row];
      if (sc > bs || (sc == bs && row < bi)) { bs = sc; bi = row; }
    }
  }
  // wave argmax (duplicated lanes carry identical (score,idx): harmless)
  for (int off = 16; off; off >>= 1) {
    float os = __shfl_xor(bs, off, 32);
    int   oi = __shfl_xor(bi, off, 32);
    if (os > bs || (os == bs && oi < bi)) { bs = os; bi = oi; }
  }
  if (lane == 0) { sSc[wave] = bs; sIx[wave] = bi; }
  __syncthreads();
  if (threadIdx.x == 0) {
    float b = sSc[0]; int i = sIx[0];
    for (int w = 1; w < 8; ++w)
      if (sSc[w] > b || (sSc[w] == b && sIx[w] < i)) { b = sSc[w]; i = sIx[w]; }
    blkScore[blockIdx.x] = b;
    blkSel[blockIdx.x]   = i;
  }
}

// Global argmax over block partials; update taken/S/c/mean incrementally.
__global__ __launch_bounds__(1024) void reduce_update(const float* __restrict__ x,
                                                      float* __restrict__ S,
                                                      float* __restrict__ mean,
                                                      float* __restrict__ c,
                                                      float* __restrict__ taken,
                                                      int*   __restrict__ idxs,
                                                      const float* __restrict__ blkScore,
                                                      const int*   __restrict__ blkSel,
                                                      int nblk, int t) {
  __shared__ float rs[1024];
  __shared__ int   ri[1024];
  __shared__ int   sSel, sWz;
  __shared__ float sC;

  float bs = -__builtin_inff(); int bi = 0x7fffffff;
  for (int bk = threadIdx.x; bk < nblk; bk += 1024) {
    float s = blkScore[bk]; int i = blkSel[bk];
    if (s > bs || (s == bs && i < bi)) { bs = s; bi = i; }
  }
  rs[threadIdx.x] = bs; ri[threadIdx.x] = bi;
  __syncthreads();
  for (int off = 512; off; off >>= 1) {
    if (threadIdx.x < off) {
      float os = rs[threadIdx.x + off]; int oi = ri[threadIdx.x + off];
      if (os > rs[threadIdx.x] || (os == rs[threadIdx.x] && oi < ri[threadIdx.x])) {
        rs[threadIdx.x] = os; ri[threadIdx.x] = oi;
      }
    }
    __syncthreads();
  }
  if (threadIdx.x == 0) {
    int sel = ri[0];
    float tk = taken[sel];
    int wz = (tk == 0.0f) ? 1 : 0;        // only fresh rows leave the mean
    taken[sel] = tk - 100000.0f;
    idxs[t] = sel;
    float cOld = *c;
    float cNew = wz ? (cOld - 1.0f) : cOld;
    *c = cNew;
    sSel = sel; sWz = wz; sC = cNew;
  }
  __syncthreads();
  int sel = sSel; int wz = sWz; float cNew = sC;
  int col = threadIdx.x;                  // blockDim == DIM
  float sv = S[col];
  if (wz) sv -= x[(size_t)sel * DIM + col];
  S[col] = sv;
  mean[col] = sv / cNew;
}

__global__ void sort_kernel(const int* __restrict__ idxs, int* __restrict__ sorted) {
  if (threadIdx.x == 0 && blockIdx.x == 0) {
    int a[ROUTE];
    for (int i = 0; i < ROUTE; ++i) a[i] = idxs[i];
    for (int i = 1; i < ROUTE; ++i) {
      int v = a[i]; int j = i - 1;
      while (j >= 0 && a[j] > v) { a[j + 1] = a[j]; --j; }
      a[j + 1] = v;
    }
    for (int i = 0; i < ROUTE; ++i) sorted[i] = a[i];
  }
}

__global__ __launch_bounds__(256) void gather_kernel(const float* __restrict__ x,
                                                     const int* __restrict__ sorted,
                                                     float* __restrict__ out) {
  __shared__ int srow;
  if (threadIdx.x == 0) srow = sorted[blockIdx.x];
  __syncthreads();
  const float4* src = (const float4*)(x + (size_t)srow * DIM);
  float4* dst = (float4*)(out + (size_t)blockIdx.x * DIM);
  for (int j = threadIdx.x; j < DIM / 4; j += 256) dst[j] = src[j];
}

extern "C" void kernel_launch(void* const* d_in, const int* in_sizes, int n_in,
                              void* d_out, int out_size, void* d_ws, size_t ws_size,
                              hipStream_t stream) {
  (void)in_sizes; (void)n_in; (void)out_size; (void)ws_size; // needs ~1.3 MB ws
  const float* x = (const float*)d_in[0];
  float* out = (float*)d_out;
  char* ws = (char*)d_ws;
  float* taken  = (float*)(ws + TAKEN_OFF);
  float* S      = (float*)(ws + S_OFF);
  float* mean   = (float*)(ws + MEAN_OFF);
  float* c      = (float*)(ws + C_OFF);
  float* part   = (float*)(ws + PART_OFF);
  float* bScore = (float*)(ws + BSCORE_OFF);
  int*   bSel   = (int*)  (ws + BIDX_OFF);
  int*   idxs   = (int*)  (ws + IDXS_OFF);
  int*   sorted = (int*)  (ws + SORT_OFF);

  init_taken<<<(NROWS + 255) / 256, 256, 0, stream>>>(taken);
  colsum_partial<<<CHUNKS * 4, 256, 0, stream>>>(x, part);
  colsum_final<<<DIM / 256, 256, 0, stream>>>(part, S, mean, c);
  for (int t = 0; t < ROUTE; ++t) {
    score_kernel<<<SCORE_BLOCKS, 256, 0, stream>>>(x, mean, taken, bScore, bSel);
    reduce_update<<<1, 1024, 0, stream>>>(x, S, mean, c, taken, idxs,
                                          bScore, bSel, SCORE_BLOCKS, t);
  }
  sort_kernel<<<1, 64, 0, stream>>>(idxs, sorted);
  gather_kernel<<<ROUTE, 256, 0, stream>>>(x, sorted, out);
}